// TOP_44066364457159
// MI455X (gfx1250) — compile-verified
//
#include <hip/hip_runtime.h>
#include <hip/hip_bf16.h>

// ---------------------------------------------------------------------------
// CDNA5 / gfx1250 implementation.
//
// Outputs (concatenated flat in d_out, reference return order):
//   [0,            2000)              new_edge_index (2 x 1000), written as float
//   [2000,         2000+2E)           edge_score (2E = 1M fp32)
//   [2000+2E,      2000+4E)           edge_label (2E fp32: 1 for pos, 0 for neg)
// ---------------------------------------------------------------------------

typedef __attribute__((ext_vector_type(2))) float v2f;
typedef __attribute__((ext_vector_type(8))) float v8f;

#define NEW_EDGES 1000
#define FEAT      128

// One wave computes 16 edge dot-products via V_WMMA_F32_16X16X4_F32:
//   A (16x4)  = 16 gathered x_i rows, 4-wide k slice
//   B (4x16)  = same k slice of 16 gathered x_j rows (column major per lane)
//   D = A*B accumulated over 32 k-steps; diag(D) = per-edge dots.
//
// fp32 16x4 A layout (ISA 7.12.2): lanes 0-15 hold {K=0,K=1} of row M=lane,
// lanes 16-31 hold {K=2,K=3} of row M=lane-16 -> each lane loads one
// contiguous float2 from its row at column kk + 2*(lane>>4). B mirrors this
// with N=lane&15. So both matrices come in as one global_load_b64 per lane
// per k-step, and every element of the gathered rows is loaded exactly once.
__global__ void __launch_bounds__(128)
edge_score_kernel(const float* __restrict__ x,
                  const int*   __restrict__ edge_index,      // (2,E) flat
                  const int*   __restrict__ neg_edge_index,  // (2,E) flat
                  const float* __restrict__ r_scalings,      // (5,)
                  const float* __restrict__ r_biases,        // (5,)
                  float*       __restrict__ out_score,       // len 2E
                  float*       __restrict__ out_label,       // len 2E
                  int E, int total)                          // total = 2E
{
    const int lane  = threadIdx.x & 31;
    const int wid   = threadIdx.x >> 5;          // wave in block (4 waves)
    const int m     = lane & 15;                 // matrix row / column index
    const int khalf = lane >> 4;                 // 0: K=0,1  1: K=2,3

    const int edge_base = (blockIdx.x * (blockDim.x >> 5) + wid) * 16;
    int e  = edge_base + m;
    int ec = (e < total) ? e : (total - 1);      // clamp tail, keep EXEC full

    // concat(edge_index, neg_edge_index): row 0 = source j, row 1 = dest i
    const int* tab = (ec < E) ? edge_index : neg_edge_index;
    const int  ee  = (ec < E) ? ec : (ec - E);
    const int  j   = tab[ee];                    // x_j row
    const int  i   = tab[ee + E];                // x_i row

    const float* pA = x + (size_t)i * FEAT;      // A rows <- x_i
    const float* pB = x + (size_t)j * FEAT;      // B cols <- x_j
    __builtin_prefetch(pA, 0, 0);                // global_prefetch_b8
    __builtin_prefetch(pB, 0, 0);

    v8f acc = {};
#pragma unroll
    for (int kk = 0; kk < FEAT; kk += 4) {
        const int col = kk + (khalf << 1);
        const v2f a = *(const v2f*)(pA + col);
        const v2f b = *(const v2f*)(pB + col);
        // 8 args: (neg_a, A, neg_b, B, c_mod, C, reuse_a, reuse_b)
        acc = __builtin_amdgcn_wmma_f32_16x16x4_f32(
            false, a, false, b, (short)0, acc, false, false);
    }

    // Diagonal extraction (16x16 fp32 C/D layout):
    //   diag[m], m<8  -> lane m,     VGPR m
    //   diag[m], m>=8 -> lane 16+m,  VGPR m-8
    const bool holds = (lane < 8) || (lane >= 24);
    const int  dm    = (lane < 8) ? lane : (lane - 16);   // which edge 0..15
    const int  sel   = (lane < 8) ? lane : (lane - 24);   // which component
    float d = acc[0];
#pragma unroll
    for (int r = 1; r < 8; ++r) d = (sel == r) ? acc[r] : d;

    // 5x: score = s[k]*elu(score) + b[k]
#pragma unroll
    for (int k = 0; k < 5; ++k) {
        const float eu = (d > 0.0f) ? d : (__expf(d) - 1.0f);
        d = r_scalings[k] * eu + r_biases[k];
    }

    const int eo = edge_base + dm;
    if (holds && eo < total) {
        out_score[eo] = d;
        out_label[eo] = (eo < E) ? 1.0f : 0.0f;
    }
}

// Rank-select instead of argsort:
//   order = argsort(-neg_score) (stable)  =>  sorted_j[p] = t0[rank(p)] where
//   rank(p) = #{q: s_q > s_p} + #{q: s_q == s_p && q < p}.
// Only p < NEW_EDGES are emitted, so 1000 blocks each scan the 2 MB
// (L2-resident) negative-score array and count.
__global__ void __launch_bounds__(256)
topk_rank_kernel(const float* __restrict__ neg_scores,     // len E (final)
                 const int*   __restrict__ neg_edge_index, // (2,E) flat
                 float*       __restrict__ out_idx,        // len 2*NEW_EDGES
                 int E)
{
    __shared__ int red[256];
    const int p  = blockIdx.x;
    const float sp = neg_scores[p];

    int cnt = 0;
    for (int q = threadIdx.x; q < E; q += blockDim.x) {
        const float sq = neg_scores[q];
        cnt += (sq > sp) || ((sq == sp) && (q < p));
    }
    red[threadIdx.x] = cnt;
    __syncthreads();
    for (int s = blockDim.x >> 1; s > 0; s >>= 1) {
        if (threadIdx.x < s) red[threadIdx.x] += red[threadIdx.x + s];
        __syncthreads();
    }
    if (threadIdx.x == 0) {
        const int r = red[0];
        out_idx[p]             = (float)neg_edge_index[r];       // t0[rank]
        out_idx[NEW_EDGES + p] = (float)neg_edge_index[E + r];   // t1[rank]
    }
}

extern "C" void kernel_launch(void* const* d_in, const int* in_sizes, int n_in,
                              void* d_out, int out_size, void* d_ws, size_t ws_size,
                              hipStream_t stream) {
    const float* x   = (const float*)d_in[0];
    const int*   ei  = (const int*)  d_in[1];
    const int*   nei = (const int*)  d_in[2];
    const float* rs  = (const float*)d_in[3];
    const float* rb  = (const float*)d_in[4];
    float* out = (float*)d_out;

    const int E     = in_sizes[1] / 2;   // 500000
    const int total = 2 * E;             // 1000000

    float* out_idx   = out;                       // 2*NEW_EDGES
    float* out_score = out + 2 * NEW_EDGES;       // total
    float* out_label = out_score + total;         // total

    // 128 threads = 4 waves, 16 edges/wave -> 64 edges per block
    const int edges_per_block = 64;
    const int blocks = (total + edges_per_block - 1) / edges_per_block;
    edge_score_kernel<<<blocks, 128, 0, stream>>>(
        x, ei, nei, rs, rb, out_score, out_label, E, total);

    // negative scores live at out_score + E
    topk_rank_kernel<<<NEW_EDGES, 256, 0, stream>>>(
        out_score + E, nei, out_idx, E);
}